// LookaheadConvolution_52707838657061
// MI455X (gfx1250) — compile-verified
//
#include <hip/hip_runtime.h>

// Problem constants (from reference)
#define SEQ   2048
#define BATCH 32
#define FEAT  1280
#define KTAPS 81            // CONTEXT + 1

// Tiling
#define T_TILE 32           // output timesteps per block (2 WMMA M-tiles)
#define B_TILE 16           // batches per block  (WMMA N)
#define F_TILE 16           // features per block (= waves per block)
#define KWIN   96           // reduction window per 16-row M-tile (16 + 80)
#define NSTEP  (KWIN / 4)   // 24 wmma 16x16x4 k-steps
#define SLAB_T (T_TILE - 16 + KWIN)   // 112 staged input rows

// x-slab LDS layout: 16B-aligned rows for async b128 fill
#define XBSTR  20                     // batch stride (floats) = 80B, 16B aligned
#define XTSTR  (B_TILE * XBSTR)       // 320 floats per t-row
// D-staging LDS layout (b32 traffic only, keep odd pad)
#define SBSTR  17
#define STSTR  (B_TILE * SBSTR)       // 272
#define NTHREADS 512

typedef __attribute__((ext_vector_type(2))) float v2f;
typedef __attribute__((ext_vector_type(8))) float v8f;
typedef __attribute__((ext_vector_type(4))) int  v4i;

typedef __attribute__((address_space(1))) v4i* gv4i_p;   // global (AS1)
typedef __attribute__((address_space(3))) v4i* lv4i_p;   // LDS    (AS3)

__global__ __launch_bounds__(NTHREADS)
void lookahead_conv_wmma(const float* __restrict__ x,
                         const float* __restrict__ w,
                         float* __restrict__ out)
{
    __shared__ float xs[SLAB_T * XTSTR];     // x slab [t][b][f], 140 KB
    __shared__ float st[16 * STSTR];         // D staging [j][b][f], 17 KB
    float* wl = st;                          // weights [16][81] alias (dead after A build)

    const int tid  = threadIdx.x;
    const int lane = tid & 31;
    const int wv   = tid >> 5;                 // wave id == local feature
    const int half = lane >> 4;                // 0: lanes 0-15, 1: lanes 16-31
    const int l15  = lane & 15;

    const int t0 = blockIdx.x * T_TILE;
    const int b0 = blockIdx.y * B_TILE;
    const int f0 = blockIdx.z * F_TILE;

    // ---- cooperative load: weights for this block's 16 features ----
    for (int i = tid; i < F_TILE * KTAPS; i += NTHREADS) {
        int fl = i / KTAPS;
        int k  = i - fl * KTAPS;
        wl[fl * KTAPS + k] = w[(f0 + fl) * KTAPS + k];
    }

    // ---- async fill: x slab [t0, t0+112) x [b0,b0+16) x [f0,f0+16) ----
    // One 16B global->LDS async move per (t, b, f-quad); zero-fill rows past SEQ.
    for (int i = tid; i < SLAB_T * B_TILE * 4; i += NTHREADS) {
        int f4 = i & 3;
        int bb = (i >> 2) & 15;
        int t  = i >> 6;
        int tg = t0 + t;
        float* d = &xs[t * XTSTR + bb * XBSTR + 4 * f4];
        if (tg < SEQ) {
            const float* g = x + ((size_t)tg * BATCH + (b0 + bb)) * FEAT + f0 + 4 * f4;
            __builtin_amdgcn_global_load_async_to_lds_b128(
                (gv4i_p)(v4i*)g, (lv4i_p)(v4i*)d, /*offset=*/0, /*cpol=*/0);
        } else {
            d[0] = 0.f; d[1] = 0.f; d[2] = 0.f; d[3] = 0.f;
        }
    }
    asm volatile("s_wait_asynccnt 0x0" ::: "memory");
    __syncthreads();

    // ---- build Toeplitz A fragments once per wave: A[j,kg] = w[f, kg - j] ----
    // 16x4 f32 A layout: lanes 0-15 hold M=j, VGPR0=K0/VGPR1=K1; lanes 16-31 K2/K3.
    v2f a[NSTEP];
    #pragma unroll
    for (int s = 0; s < NSTEP; ++s) {
        int i0 = 4 * s + 2 * half - l15;      // kg - j for this lane, reg 0
        int i1 = i0 + 1;                      // reg 1
        float a0 = (i0 >= 0 && i0 < KTAPS) ? wl[wv * KTAPS + i0] : 0.f;
        float a1 = (i1 >= 0 && i1 < KTAPS) ? wl[wv * KTAPS + i1] : 0.f;
        a[s].x = a0;
        a[s].y = a1;
    }
    __syncthreads();   // wl (aliased with st) must be fully read before st is written

    // ---- two 16x16 output tiles (time) per block ----
    #pragma unroll
    for (int mt = 0; mt < T_TILE / 16; ++mt) {
        const int ttl = mt * 16;

        v8f acc = {};
        // B (4x16 f32) layout: lanes 0-15 rows K0(VGPR0)/K1(VGPR1); lanes 16-31 K2/K3.
        #pragma unroll
        for (int s = 0; s < NSTEP; ++s) {
            int tb = ttl + 4 * s + 2 * half;
            v2f bfrag;
            bfrag.x = xs[(tb + 0) * XTSTR + l15 * XBSTR + wv];
            bfrag.y = xs[(tb + 1) * XTSTR + l15 * XBSTR + wv];
            acc = __builtin_amdgcn_wmma_f32_16x16x4_f32(
                      /*neg_a=*/false, a[s],
                      /*neg_b=*/false, bfrag,
                      /*c_mod=*/(short)0, acc,
                      /*reuse_a=*/false, /*reuse_b=*/false);
        }

        // stage D tile into LDS: VGPR v -> row j = v + 8*half, col = l15, feat = wv
        #pragma unroll
        for (int v = 0; v < 8; ++v) {
            int j = v + 8 * half;
            st[j * STSTR + l15 * SBSTR + wv] = acc[v];
        }
        __syncthreads();

        // cooperative coalesced store along contiguous FEAT
        for (int i = tid; i < 16 * B_TILE * 4; i += NTHREADS) {
            int f4 = i & 3;
            int bb = (i >> 2) & 15;
            int j  = i >> 6;
            const float* sp = &st[j * STSTR + bb * SBSTR + 4 * f4];
            float4 v = make_float4(sp[0], sp[1], sp[2], sp[3]);
            *(float4*)(out + ((size_t)(t0 + ttl + j) * BATCH + (b0 + bb)) * FEAT
                           + f0 + 4 * f4) = v;
        }
        __syncthreads();
    }
}

extern "C" void kernel_launch(void* const* d_in, const int* in_sizes, int n_in,
                              void* d_out, int out_size, void* d_ws, size_t ws_size,
                              hipStream_t stream) {
    const float* x = (const float*)d_in[0];       // (SEQ, BATCH, FEAT) fp32
    const float* w = (const float*)d_in[1];       // (FEAT, KTAPS)      fp32
    float* out = (float*)d_out;                   // (SEQ, BATCH, FEAT) fp32

    dim3 grid(SEQ / T_TILE, BATCH / B_TILE, FEAT / F_TILE);   // 64 x 2 x 80
    lookahead_conv_wmma<<<grid, NTHREADS, 0, stream>>>(x, w, out);
}